// ResidualGCNLayer_53068615909524
// MI455X (gfx1250) — compile-verified
//
#include <hip/hip_runtime.h>

#define DD   128
#define EPSF 1e-5f

typedef float v2f __attribute__((ext_vector_type(2)));
typedef float v4f __attribute__((ext_vector_type(4)));
typedef float v8f __attribute__((ext_vector_type(8)));

// ---------------- zero helpers ----------------
__global__ void k_zero4(v4f* __restrict__ p, long n4) {
    long i = (long)blockIdx.x * blockDim.x + threadIdx.x;
    long stride = (long)gridDim.x * blockDim.x;
    v4f z = {0.f, 0.f, 0.f, 0.f};
    for (; i < n4; i += stride) p[i] = z;
}

__global__ void k_zero(float* __restrict__ p, int n) {
    int i = blockIdx.x * blockDim.x + threadIdx.x;
    if (i < n) p[i] = 0.f;
}

// ---------------- edge scatter-add: agg[dst] += feats[src] ----------------
// 32 lanes per edge, float4 per lane (128 ch). feats & agg are L2-resident (51 MB each),
// so this phase is bounded by L2 atomic throughput, not HBM.
__global__ __launch_bounds__(256)
void k_scatter(const float* __restrict__ feats, const int* __restrict__ src,
               const int* __restrict__ dst, float* __restrict__ agg, int E) {
    long idx = (long)blockIdx.x * blockDim.x + threadIdx.x;
    int e = (int)(idx >> 5);
    int l = (int)(idx & 31);
    if (e < E) {
        int s = src[e];
        int d = dst[e];
        v4f v = *(const v4f*)(feats + (long)s * DD + l * 4);
        float* o = agg + (long)d * DD + l * 4;
        unsafeAtomicAdd(o + 0, v.x);
        unsafeAtomicAdd(o + 1, v.y);
        unsafeAtomicAdd(o + 2, v.z);
        unsafeAtomicAdd(o + 3, v.w);
    }
}

// ---------------- fused dual GEMM + bias + ReLU + add + BN1 stats ----------------
// t = relu(agg@Wg + bg) + relu(feats@Wr + br); per-channel sum/sumsq accumulated.
// One wave owns 16 rows. V_WMMA_F32_16X16X4_F32; K loop over D=128, k-outer over
// groups of 4 column tiles (8 live v8f accumulators) so A fragments load once per group.
// Fragment layouts (ISA 7.12.2, 32-bit):
//   A 16x4 : lane<16 -> {A[m][k], A[m][k+1]}, lane>=16 -> {A[m][k+2], A[m][k+3]}
//   B 4x16 : lane<16 -> {B[k][n], B[k+1][n]}, lane>=16 -> {B[k+2][n], B[k+3][n]}
//   C/D    : reg r, lane<16 -> (M=r, N=lane); lane>=16 -> (M=r+8, N=lane-16)
__global__ __launch_bounds__(256)
void k_gemm_stats(const float* __restrict__ agg, const float* __restrict__ feats,
                  const float* __restrict__ Wg, const float* __restrict__ bg,
                  const float* __restrict__ Wr, const float* __restrict__ br,
                  float* __restrict__ t, float* __restrict__ sum_t,
                  float* __restrict__ ssq_t, int n) {
    __shared__ float lsum[DD];
    __shared__ float lssq[DD];
    if (threadIdx.x < DD) { lsum[threadIdx.x] = 0.f; lssq[threadIdx.x] = 0.f; }
    __syncthreads();

    const int lane = threadIdx.x & 31;
    const int wave = threadIdx.x >> 5;
    const int half = lane >> 4;      // K-half / M-half selector
    const int lm   = lane & 15;
    const int m0   = blockIdx.x * 128 + wave * 16;
    const bool full = (m0 + 16 <= n);   // wave-uniform: fast path has no store masking

    int arow = m0 + lm;
    if (arow > n - 1) arow = n - 1;  // clamp tail loads; stores masked below
    const float* aAgg = agg   + (long)arow * DD + 2 * half;
    const float* aFt  = feats + (long)arow * DD + 2 * half;

    for (int g = 0; g < 2; ++g) {        // two groups of 4 column tiles
        v8f acc[8];
#pragma unroll
        for (int q = 0; q < 8; ++q) acc[q] = (v8f){};

        for (int k = 0; k < DD; k += 4) {
            v2f a0 = *(const v2f*)(aAgg + k);
            v2f a1 = *(const v2f*)(aFt + k);
            const int kr = k + 2 * half;
#pragma unroll
            for (int ti = 0; ti < 4; ++ti) {
                const int col = (g * 4 + ti) * 16 + lm;
                v2f b0 = { Wg[kr * DD + col], Wg[(kr + 1) * DD + col] };
                v2f b1 = { Wr[kr * DD + col], Wr[(kr + 1) * DD + col] };
                acc[2 * ti] = __builtin_amdgcn_wmma_f32_16x16x4_f32(
                    false, a0, false, b0, (short)0, acc[2 * ti], false, false);
                acc[2 * ti + 1] = __builtin_amdgcn_wmma_f32_16x16x4_f32(
                    false, a1, false, b1, (short)0, acc[2 * ti + 1], false, false);
            }
        }

#pragma unroll
        for (int ti = 0; ti < 4; ++ti) {
            const int col = (g * 4 + ti) * 16 + lm;
            const float bgv = bg[col];
            const float brv = br[col];
            float ps = 0.f, pq = 0.f;
            if (full) {
#pragma unroll
                for (int r = 0; r < 8; ++r) {
                    const int m = m0 + r + 8 * half;
                    float hv = fmaxf(acc[2 * ti][r] + bgv, 0.f) +
                               fmaxf(acc[2 * ti + 1][r] + brv, 0.f);
                    t[(long)m * DD + col] = hv;
                    ps += hv;
                    pq += hv * hv;
                }
            } else {
#pragma unroll
                for (int r = 0; r < 8; ++r) {
                    const int m = m0 + r + 8 * half;
                    float hv = fmaxf(acc[2 * ti][r] + bgv, 0.f) +
                               fmaxf(acc[2 * ti + 1][r] + brv, 0.f);
                    if (m < n) {
                        t[(long)m * DD + col] = hv;
                        ps += hv;
                        pq += hv * hv;
                    }
                }
            }
            atomicAdd(&lsum[col], ps);   // ds_add_f32
            atomicAdd(&lssq[col], pq);
        }
    }
    __syncthreads();
    if (threadIdx.x < DD) {
        unsafeAtomicAdd(sum_t + threadIdx.x, lsum[threadIdx.x]);
        unsafeAtomicAdd(ssq_t + threadIdx.x, lssq[threadIdx.x]);
    }
}

// ---------------- finalize BN stats: mu = sum/N, inv = rsqrt(E[x^2]-mu^2+eps) ----------------
__global__ void k_finalize(const float* __restrict__ sum, const float* __restrict__ ssq,
                           float* __restrict__ mu, float* __restrict__ inv, float invN) {
    int c = threadIdx.x;
    if (c < DD) {
        float m = sum[c] * invN;
        float v = ssq[c] * invN - m * m;
        mu[c] = m;
        inv[c] = rsqrtf(v + EPSF);
    }
}

// ---------------- s = BN1(t) + feats, accumulate BN2 stats ----------------
__global__ __launch_bounds__(256)
void k_bn1_add(const float* __restrict__ t, const float* __restrict__ feats,
               const float* __restrict__ g1, const float* __restrict__ be1,
               const float* __restrict__ mu1, const float* __restrict__ inv1,
               float* __restrict__ s, float* __restrict__ sum_s,
               float* __restrict__ ssq_s, long n4) {
    __shared__ float lsum[DD];
    __shared__ float lssq[DD];
    if (threadIdx.x < DD) { lsum[threadIdx.x] = 0.f; lssq[threadIdx.x] = 0.f; }
    __syncthreads();

    long i = (long)blockIdx.x * blockDim.x + threadIdx.x;   // float4 index
    const int c0 = ((int)(i & 31)) * 4;                     // channel group (row = 32 f4)
    float ps[4] = {0.f, 0.f, 0.f, 0.f};
    float pq[4] = {0.f, 0.f, 0.f, 0.f};
    if (i < n4) {
        v4f tv = ((const v4f*)t)[i];
        v4f fv = ((const v4f*)feats)[i];
        v4f sv;
#pragma unroll
        for (int j = 0; j < 4; ++j) {
            const int c = c0 + j;
            float val = g1[c] * (tv[j] - mu1[c]) * inv1[c] + be1[c] + fv[j];
            sv[j] = val;
            ps[j] = val;
            pq[j] = val * val;
        }
        ((v4f*)s)[i] = sv;
    }
#pragma unroll
    for (int j = 0; j < 4; ++j) {
        atomicAdd(&lsum[c0 + j], ps[j]);
        atomicAdd(&lssq[c0 + j], pq[j]);
    }
    __syncthreads();
    if (threadIdx.x < DD) {
        unsafeAtomicAdd(sum_s + threadIdx.x, lsum[threadIdx.x]);
        unsafeAtomicAdd(ssq_s + threadIdx.x, lssq[threadIdx.x]);
    }
}

// ---------------- out = BN2(s) ----------------
__global__ __launch_bounds__(256)
void k_bn2(const float* __restrict__ s, const float* __restrict__ g2,
           const float* __restrict__ be2, const float* __restrict__ mu2,
           const float* __restrict__ inv2, float* __restrict__ out, long n4) {
    long i = (long)blockIdx.x * blockDim.x + threadIdx.x;
    if (i < n4) {
        const int c0 = ((int)(i & 31)) * 4;
        v4f sv = ((const v4f*)s)[i];
        v4f ov;
#pragma unroll
        for (int j = 0; j < 4; ++j) {
            const int c = c0 + j;
            ov[j] = g2[c] * (sv[j] - mu2[c]) * inv2[c] + be2[c];
        }
        ((v4f*)out)[i] = ov;
    }
}

extern "C" void kernel_launch(void* const* d_in, const int* in_sizes, int n_in,
                              void* d_out, int out_size, void* d_ws, size_t ws_size,
                              hipStream_t stream) {
    const float* feats = (const float*)d_in[0];
    const float* Wg    = (const float*)d_in[1];
    const float* bg    = (const float*)d_in[2];
    const float* Wr    = (const float*)d_in[3];
    const float* br    = (const float*)d_in[4];
    const float* g1    = (const float*)d_in[5];
    const float* be1   = (const float*)d_in[6];
    const float* g2    = (const float*)d_in[7];
    const float* be2   = (const float*)d_in[8];
    const int*   src   = (const int*)d_in[9];
    const int*   dst   = (const int*)d_in[10];

    const int n = in_sizes[0] / DD;   // 100000
    const int E = in_sizes[9];        // 1600000
    const long nd  = (long)n * DD;
    const long n4  = nd / 4;

    // workspace: [agg | t | stats]; agg region is reused for s after the GEMM pass
    float* agg   = (float*)d_ws;
    float* t     = agg + nd;
    float* stats = t + nd;
    float* sum_t = stats + 0,   *ssq_t = stats + 128;
    float* mu1   = stats + 256, *inv1  = stats + 384;
    float* sum_s = stats + 512, *ssq_s = stats + 640;
    float* mu2   = stats + 768, *inv2  = stats + 896;
    float* s     = agg;  // reuse

    // 1) zero agg + stat accumulators (every call: deterministic under graph replay)
    k_zero4<<<4096, 256, 0, stream>>>((v4f*)agg, n4);
    k_zero<<<4, 256, 0, stream>>>(stats, 1024);

    // 2) edge aggregation (L2-resident scatter-add)
    {
        long threads = (long)E * 32;
        int grid = (int)((threads + 255) / 256);
        k_scatter<<<grid, 256, 0, stream>>>(feats, src, dst, agg, E);
    }

    // 3) dual WMMA GEMM + bias + relu + add, BN1 stats
    {
        int grid = (n + 127) / 128;
        k_gemm_stats<<<grid, 256, 0, stream>>>(agg, feats, Wg, bg, Wr, br,
                                               t, sum_t, ssq_t, n);
    }

    // 4) BN1 finalize
    k_finalize<<<1, 128, 0, stream>>>(sum_t, ssq_t, mu1, inv1, 1.0f / (float)n);

    // 5) s = BN1(t) + feats, BN2 stats
    {
        int grid = (int)((n4 + 255) / 256);
        k_bn1_add<<<grid, 256, 0, stream>>>(t, feats, g1, be1, mu1, inv1,
                                            s, sum_s, ssq_s, n4);
    }

    // 6) BN2 finalize
    k_finalize<<<1, 128, 0, stream>>>(sum_s, ssq_s, mu2, inv2, 1.0f / (float)n);

    // 7) out = BN2(s)
    {
        int grid = (int)((n4 + 255) / 256);
        k_bn2<<<grid, 256, 0, stream>>>(s, g2, be2, mu2, inv2, (float*)d_out, n4);
    }
}